// binary_decoder_19628000543520
// MI455X (gfx1250) — compile-verified
//
#include <hip/hip_runtime.h>
#include <math.h>

#define B_SZ 8
#define F_SZ 2048
#define O_SZ 512
#define NB 8
#define N_SZ (O_SZ * NB)   // 4096
#define THRESH 0.5f

#define KSPLIT 4
#define KCHUNK (F_SZ / KSPLIT)   // 512

typedef __attribute__((ext_vector_type(2))) float v2f;
typedef __attribute__((ext_vector_type(8))) float v8f;

// ---------------------------------------------------------------------------
// Pass 1: partial[kc][b][n] = sum_{f in chunk kc} x[b,f] * (w[f,n] >= 0.5)
// FP32 WMMA 16x16x4 (exact x*bit products; the mod-2 epilogue is
// bit-sensitive, so no low-precision WMMA flavors).
// Grid = 32 N-blocks x 4 K-chunks -> 1024 waves, enough concurrency to pull
// the 32 MB read-once weight stream at the 23.3 TB/s roofline (~1.4 us).
// Weight loads are non-temporal (TH=NT) to avoid polluting WGP$/L2.
// x slice (16 KB) staged in LDS column-major (conflict-free A-frag reads).
// ---------------------------------------------------------------------------
__global__ __launch_bounds__(256) void colsum_wmma_f32(const float* __restrict__ x,
                                                       const float* __restrict__ w,
                                                       float* __restrict__ partial) {
  const int kc = blockIdx.y;           // K-chunk index
  const int k0 = kc * KCHUNK;

  // x chunk staged column-major: xs[f*8 + b], f local to chunk
  __shared__ float xs[KCHUNK * B_SZ];

  for (int i = threadIdx.x; i < B_SZ * KCHUNK; i += 256) {
    const int b = i >> 9;               // i / KCHUNK
    const int f = i & (KCHUNK - 1);     // i % KCHUNK
    xs[f * B_SZ + b] = x[(size_t)b * F_SZ + k0 + f];
  }
  __syncthreads();

  const int lane = threadIdx.x & 31;
  const int wave = threadIdx.x >> 5;
  const int l    = lane & 15;      // N index (B frag) / M index (A frag)
  const int half = lane >> 4;      // selects K pair {0,1} vs {2,3}
  const int krow = 2 * half;
  const int n0   = blockIdx.x * 128 + wave * 16;

  const int   row8 = l & 7;                       // valid x row for every lane
  const float msk  = (l < B_SZ) ? 1.0f : 0.0f;    // zero-pad M rows 8..15

  const float* __restrict__ wcol = w + (size_t)k0 * N_SZ + n0 + l;

  v8f acc = {};

#pragma unroll 4
  for (int k = 0; k < KCHUNK; k += 4) {
    // A-frag from LDS: A[M=l, K = krow + {0,1}] (unconditional, masked by mul)
    const float xa0 = xs[(k + krow + 0) * B_SZ + row8];
    const float xa1 = xs[(k + krow + 1) * B_SZ + row8];
    v2f a;
    a.x = xa0 * msk;
    a.y = xa1 * msk;

    // B-frag: B[K = krow + {0,1}, N = n0 + l], binarized; read-once -> NT
    const float w0 = __builtin_nontemporal_load(&wcol[(size_t)(k + krow + 0) * N_SZ]);
    const float w1 = __builtin_nontemporal_load(&wcol[(size_t)(k + krow + 1) * N_SZ]);
    v2f b;
    b.x = (w0 >= THRESH) ? 1.0f : 0.0f;
    b.y = (w1 >= THRESH) ? 1.0f : 0.0f;

    acc = __builtin_amdgcn_wmma_f32_16x16x4_f32(
        /*neg_a=*/false, a, /*neg_b=*/false, b,
        /*c_mod=*/(short)0, acc, /*reuse_a=*/false, /*reuse_b=*/false);
  }

  // C/D layout: VGPR r, lanes 0-15 -> M=r (valid batches 0..7), N = lane
  float* __restrict__ part = partial + (size_t)kc * B_SZ * N_SZ;
  if (lane < 16) {
#pragma unroll
    for (int r = 0; r < B_SZ; ++r)
      part[(size_t)r * N_SZ + n0 + lane] = acc[r];
  }
}

// ---------------------------------------------------------------------------
// Pass 2: closed-form carry-save-adder reconstruction.
//   s_j(f)  = (A_j(f) + H_{j-1}(f-1)) mod 2
//   H_j(f)  = (A_j(f) + H_{j-1}(f-1) - s_j(f)) / 2
//   c_out_j = H_{j-1}(F-1) - H_{j-1}(F-2),  c_out_0 = 0
// Needs A_j at the last 9 staggered offsets -> full sum (reduced over the 4
// K-split partials) minus the last <=8 row contributions, computed directly.
// ---------------------------------------------------------------------------
__global__ __launch_bounds__(256) void csa_reconstruct(const float* __restrict__ x,
                                                       const float* __restrict__ w,
                                                       const float* __restrict__ partial,
                                                       float* __restrict__ out) {
  const int m = blockIdx.x * blockDim.x + threadIdx.x;
  if (m >= B_SZ * O_SZ) return;
  const int b = m / O_SZ;
  const int o = m % O_SZ;

  // Ap[j][d] = A_j(F-1-d), d = 0..8
  float Ap[NB][9];
#pragma unroll
  for (int j = 0; j < NB; ++j) {
    float s = 0.0f;
#pragma unroll
    for (int kc = 0; kc < KSPLIT; ++kc)
      s += partial[((size_t)kc * B_SZ + b) * N_SZ + o * NB + j];
    Ap[j][0] = s;
  }

#pragma unroll
  for (int d = 1; d <= 8; ++d) {
    const int f = F_SZ - d;
    const float xv = x[(size_t)b * F_SZ + f];
#pragma unroll
    for (int j = 0; j < NB; ++j) {
      const float wv = w[(size_t)f * N_SZ + o * NB + j];
      const float a = (wv >= THRESH) ? xv : 0.0f;
      Ap[j][d] = Ap[j][d - 1] - a;
    }
  }

  float Hprev[10], Hcur[10];
#pragma unroll
  for (int d = 0; d < 10; ++d) { Hprev[d] = 0.0f; Hcur[d] = 0.0f; }

  float s_out[NB], c_out[NB];
  c_out[0] = 0.0f;

#pragma unroll
  for (int j = 0; j < NB; ++j) {
#pragma unroll
    for (int d = 0; d < 9; ++d) {
      if (d <= 8 - j) {
        const float t = Ap[j][d] + Hprev[d + 1];
        const float s = t - 2.0f * floorf(t * 0.5f);
        Hcur[d] = (t - s) * 0.5f;
        if (d == 0) s_out[j] = s;
      }
    }
    if (j + 1 < NB) c_out[j + 1] = Hcur[0] - Hcur[1];
#pragma unroll
    for (int d = 0; d < 10; ++d) Hprev[d] = Hcur[d];
  }

  // s -> [B, NB*O] at out[0..32767]; carry -> [B*O, NB] at out[32768..]
#pragma unroll
  for (int j = 0; j < NB; ++j) {
    out[(size_t)b * N_SZ + o * NB + j] = s_out[j];
    out[(size_t)B_SZ * N_SZ + (size_t)m * NB + j] = c_out[j];
  }
}

extern "C" void kernel_launch(void* const* d_in, const int* in_sizes, int n_in,
                              void* d_out, int out_size, void* d_ws, size_t ws_size,
                              hipStream_t stream) {
  const float* x = (const float*)d_in[0];   // [8, 2048] f32
  const float* w = (const float*)d_in[1];   // [2048, 4096] f32
  float* out     = (float*)d_out;           // 65536 f32 (s ++ carry)
  float* partial = (float*)d_ws;            // KSPLIT * 8 * 4096 f32 = 512 KB

  colsum_wmma_f32<<<dim3(N_SZ / 128, KSPLIT), dim3(256), 0, stream>>>(x, w, partial);
  csa_reconstruct<<<dim3((B_SZ * O_SZ + 255) / 256), dim3(256), 0, stream>>>(x, w, partial, out);
}